// Generator_35527969472799
// MI455X (gfx1250) — compile-verified
//
#include <hip/hip_runtime.h>
#include <math.h>

typedef __attribute__((ext_vector_type(2))) float v2f;
typedef __attribute__((ext_vector_type(8))) float v8f;

#define HP 256
#define WP 256
#define HE 2048
#define WE 4096
#define BATCH 64
#define NPIX (HP * WP)
#define TILES 16                         // stats blocks per batch
#define PIX_PER_BLOCK (NPIX / TILES)     // 4096
#define STAT_ITERS (PIX_PER_BLOCK / (256 * 2))  // 8: 256 thr x 2 px / iter

// workspace layout (in floats)
#define WS_R     0        // 64*9 rotation matrices
#define WS_INVF  576      // 1/f
#define WS_PART  640      // 64*16*6 block partials (s0,s1,s2,q0,q1,q2)
#define WS_COEF  8192     // 64*4 (a0,a1,a2,offset)

#define PI_F 3.14159265358979323846f

// ---------------------------------------------------------------------------
// Bilinear equirect sample with horizontal/vertical wrap, 3 channels.
// ---------------------------------------------------------------------------
__device__ __forceinline__ void sample_rgb(const float* __restrict__ equi,
                                           const float R[9], float invf,
                                           int px, int py,
                                           float& v0, float& v1, float& v2)
{
    float rx = ((float)px - 128.0f) * invf;
    float ry = ((float)py - 128.0f) * invf;
    float mx = R[0] * rx + R[1] * ry + R[2];
    float my = R[3] * rx + R[4] * ry + R[5];
    float mz = R[6] * rx + R[7] * ry + R[8];
    float inorm = rsqrtf(mx * mx + my * my + mz * mz);
    float phi   = asinf(mz * inorm);
    float theta = atan2f(my, mx);
    // ui = mod(theta*2048/pi - 2047.5, 4096); uj = mod(phi*2048/pi - 1023.5, 2048)
    float u = theta * (2048.0f / PI_F) - 2047.5f;
    u = fmodf(u, 4096.0f); if (u < 0.0f) u += 4096.0f;
    float v = phi * (2048.0f / PI_F) - 1023.5f;
    v = fmodf(v, 2048.0f); if (v < 0.0f) v += 2048.0f;
    float x0f = floorf(u), y0f = floorf(v);
    float wx = u - x0f, wy = v - y0f;
    int x0 = ((int)x0f) & (WE - 1);
    int x1 = (x0 + 1)   & (WE - 1);
    int y0 = ((int)y0f) & (HE - 1);
    int y1 = (y0 + 1)   & (HE - 1);
    float w00 = (1.0f - wx) * (1.0f - wy), w01 = wx * (1.0f - wy);
    float w10 = (1.0f - wx) * wy,          w11 = wx * wy;
    int i00 = y0 * WE + x0, i01 = y0 * WE + x1;
    int i10 = y1 * WE + x0, i11 = y1 * WE + x1;
    const float* p0 = equi;
    const float* p1 = equi + HE * WE;
    const float* p2 = equi + 2 * HE * WE;
    v0 = w00 * p0[i00] + w01 * p0[i01] + w10 * p0[i10] + w11 * p0[i11];
    v1 = w00 * p1[i00] + w01 * p1[i01] + w10 * p1[i10] + w11 * p1[i11];
    v2 = w00 * p2[i00] + w01 * p2[i01] + w10 * p2[i10] + w11 * p2[i11];
}

// ---------------------------------------------------------------------------
// Kernel 1: per-batch rotation matrices + focal. R = Rz(yaw) Ry(pitch) Rx(roll)
// ---------------------------------------------------------------------------
__global__ void prep_kernel(const float* fov, const float* roll,
                            const float* pitch, const float* yaw, float* ws)
{
    int b = threadIdx.x;
    if (b == 0) {
        // f = 128 / tan(deg2rad(fov)/2)  ->  invf = tan(fov*pi/360)/128
        ws[WS_INVF] = tanf(fov[0] * (PI_F / 360.0f)) * (1.0f / 128.0f);
    }
    if (b < BATCH) {
        float cr = cosf(roll[0]),  sr = sinf(roll[0]);
        float cp = cosf(pitch[0]), sp = sinf(pitch[0]);
        float cy = cosf(yaw[b]),   sy = sinf(yaw[b]);
        float* R = ws + WS_R + b * 9;
        R[0] = cy * cp;  R[1] = cy * sp * sr - sy * cr;  R[2] = cy * sp * cr + sy * sr;
        R[3] = sy * cp;  R[4] = sy * sp * sr + cy * cr;  R[5] = sy * sp * cr - cy * sr;
        R[6] = -sp;      R[7] = cp * sr;                 R[8] = cp * cr;
    }
}

// ---------------------------------------------------------------------------
// Kernel 2: sample pass 1 — accumulate per-(batch,channel) sum and sum-of-
// squares on the matrix pipe. Ones(16x4) x B(4x16) + C: every row of C becomes
// the column sums of B, so C accumulates the wave's 64 packed values per WMMA
// regardless of B's lane layout. laneSum(C[0]) = 2 * total (rows 0 and 8).
// ---------------------------------------------------------------------------
__global__ void stats_kernel(const float* __restrict__ equi,
                             const float* wsr, float* wsw)
{
    const int batch = blockIdx.y;
    const int tile  = blockIdx.x;

    float R[9];
#pragma unroll
    for (int i = 0; i < 9; ++i) R[i] = wsr[WS_R + batch * 9 + i];
    const float invf = wsr[WS_INVF];

    v2f ones = {1.0f, 1.0f};
    v8f s0 = {}, s1 = {}, s2 = {};   // channel sums
    v8f q0 = {}, q1 = {}, q2 = {};   // channel sums of squares

    const int base = tile * PIX_PER_BLOCK;
#pragma unroll 1
    for (int it = 0; it < STAT_ITERS; ++it) {
        int p = base + it * 512 + (int)threadIdx.x * 2;   // even
        float pa0, pa1, pa2, pb0, pb1, pb2;
        sample_rgb(equi, R, invf, p & (WP - 1), p >> 8, pa0, pa1, pa2);
        sample_rgb(equi, R, invf, (p + 1) & (WP - 1), (p + 1) >> 8, pb0, pb1, pb2);

        v2f bv;
        bv.x = pa0; bv.y = pb0;
        s0 = __builtin_amdgcn_wmma_f32_16x16x4_f32(false, ones, false, bv, (short)0, s0, false, false);
        bv.x = pa1; bv.y = pb1;
        s1 = __builtin_amdgcn_wmma_f32_16x16x4_f32(false, ones, false, bv, (short)0, s1, false, false);
        bv.x = pa2; bv.y = pb2;
        s2 = __builtin_amdgcn_wmma_f32_16x16x4_f32(false, ones, false, bv, (short)0, s2, false, false);
        bv.x = pa0 * pa0; bv.y = pb0 * pb0;
        q0 = __builtin_amdgcn_wmma_f32_16x16x4_f32(false, ones, false, bv, (short)0, q0, false, false);
        bv.x = pa1 * pa1; bv.y = pb1 * pb1;
        q1 = __builtin_amdgcn_wmma_f32_16x16x4_f32(false, ones, false, bv, (short)0, q1, false, false);
        bv.x = pa2 * pa2; bv.y = pb2 * pb2;
        q2 = __builtin_amdgcn_wmma_f32_16x16x4_f32(false, ones, false, bv, (short)0, q2, false, false);
    }

    // wave-level: butterfly over the 32 copies of row0/row8 column sums
    float w[6] = { s0[0], s1[0], s2[0], q0[0], q1[0], q2[0] };
#pragma unroll
    for (int m = 16; m >= 1; m >>= 1) {
#pragma unroll
        for (int k = 0; k < 6; ++k) w[k] += __shfl_xor(w[k], m, 32);
    }

    __shared__ float red[8][6];
    const int wave = threadIdx.x >> 5;
    const int lane = threadIdx.x & 31;
    if (lane == 0) {
#pragma unroll
        for (int k = 0; k < 6; ++k) red[wave][k] = w[k] * 0.5f;  // /2: two row copies
    }
    __syncthreads();
    if (threadIdx.x == 0) {
#pragma unroll
        for (int k = 0; k < 6; ++k) {
            float t = 0.0f;
#pragma unroll
            for (int wv = 0; wv < 8; ++wv) t += red[wv][k];
            wsw[WS_PART + (batch * TILES + tile) * 6 + k] = t;
        }
    }
}

// ---------------------------------------------------------------------------
// Kernel 3: fold mean/std/gray-weights into a0..a2, offset per batch.
// gray = sum_c w_c * (v_c - mean_c)/std_c = a0 v0 + a1 v1 + a2 v2 + off
// ---------------------------------------------------------------------------
__global__ void coeffs_kernel(const float* wsr, float* wsw)
{
    int b = threadIdx.x;
    if (b >= BATCH) return;
    float s[3] = {0.f, 0.f, 0.f}, q[3] = {0.f, 0.f, 0.f};
    for (int t = 0; t < TILES; ++t) {
        const float* p = wsr + WS_PART + (b * TILES + t) * 6;
#pragma unroll
        for (int c = 0; c < 3; ++c) { s[c] += p[c]; q[c] += p[3 + c]; }
    }
    const float N = (float)NPIX;
    const float gw[3] = {0.2989f, 0.587f, 0.114f};
    float a[3], off = 0.0f;
#pragma unroll
    for (int c = 0; c < 3; ++c) {
        float mean = s[c] / N;
        float var  = (q[c] - s[c] * mean) / (N - 1.0f);  // unbiased
        float istd = rsqrtf(var);
        a[c] = gw[c] * istd;
        off -= a[c] * mean;
    }
    wsw[WS_COEF + b * 4 + 0] = a[0];
    wsw[WS_COEF + b * 4 + 1] = a[1];
    wsw[WS_COEF + b * 4 + 2] = a[2];
    wsw[WS_COEF + b * 4 + 3] = off;
}

// ---------------------------------------------------------------------------
// Kernel 4: pass 2 — resample (equi now L2-resident) and emit gray output.
// ---------------------------------------------------------------------------
__global__ void gray_kernel(const float* __restrict__ equi,
                            const float* ws, float* __restrict__ out)
{
    const int batch = blockIdx.y;
    const int p = blockIdx.x * 256 + threadIdx.x;

    float R[9];
#pragma unroll
    for (int i = 0; i < 9; ++i) R[i] = ws[WS_R + batch * 9 + i];
    const float invf = ws[WS_INVF];
    const float a0 = ws[WS_COEF + batch * 4 + 0];
    const float a1 = ws[WS_COEF + batch * 4 + 1];
    const float a2 = ws[WS_COEF + batch * 4 + 2];
    const float of = ws[WS_COEF + batch * 4 + 3];

    float v0, v1, v2;
    sample_rgb(equi, R, invf, p & (WP - 1), p >> 8, v0, v1, v2);
    out[batch * NPIX + p] = a0 * v0 + a1 * v1 + a2 * v2 + of;
}

// ---------------------------------------------------------------------------
extern "C" void kernel_launch(void* const* d_in, const int* in_sizes, int n_in,
                              void* d_out, int out_size, void* d_ws, size_t ws_size,
                              hipStream_t stream)
{
    const float* equi  = (const float*)d_in[0];
    const float* fov   = (const float*)d_in[1];
    const float* roll  = (const float*)d_in[2];
    const float* pitch = (const float*)d_in[3];
    const float* yaw   = (const float*)d_in[4];
    float* ws  = (float*)d_ws;
    float* out = (float*)d_out;

    prep_kernel<<<1, 64, 0, stream>>>(fov, roll, pitch, yaw, ws);
    stats_kernel<<<dim3(TILES, BATCH), 256, 0, stream>>>(equi, ws, ws);
    coeffs_kernel<<<1, 64, 0, stream>>>(ws, ws);
    gray_kernel<<<dim3(NPIX / 256, BATCH), 256, 0, stream>>>(equi, ws, out);
}